// OTStyleModel_10411000725761
// MI455X (gfx1250) — compile-verified
//
#include <hip/hip_runtime.h>
#include <hip/hip_bf16.h>
#include <math.h>

typedef __attribute__((ext_vector_type(2))) float v2f;
typedef __attribute__((ext_vector_type(8))) float v8f;

#define B_     512
#define T_     128
#define D_     768
#define CD_    192
#define SD_    128
#define P_     128
#define FEAT_  321
#define TWO_D  1536
#define NFUSED 320

// LDS staging for proj B-operand (dirs_n): K-chunk of 32, row stride padded
// to 36 floats (144 B: 16B-aligned rows, bank = 36*r mod 64 distinct over 16
// consecutive rows -> conflict-free fragment reads).
#define KC_    32
#define SROW_  36
#define NCHUNK (D_ / KC_)   // 24

__device__ __forceinline__ v2f ld2(const float* p) {
  const float2 t = *(const float2*)p;
  v2f r; r.x = t.x; r.y = t.y; return r;
}

// ---- gfx1250 async global->LDS copy (ASYNCcnt-tracked), builtin or asm ----
__device__ __forceinline__ void async_copy16(const float* g, float* l) {
#if __has_builtin(__builtin_amdgcn_global_load_async_to_lds_b128)
  typedef int v4i __attribute__((vector_size(16)));
  typedef __attribute__((address_space(1))) v4i gv4i;
  typedef __attribute__((address_space(3))) v4i lv4i;
  __builtin_amdgcn_global_load_async_to_lds_b128((gv4i*)g, (lv4i*)l, 0, 0);
#else
  const unsigned loff = (unsigned)(unsigned long long)l;  // LDS offset = addr[31:0]
  asm volatile("global_load_async_to_lds_b128 %0, %1, off"
               :: "v"(loff), "v"(g) : "memory");
#endif
}

__device__ __forceinline__ void wait_asynccnt0() {
#if __has_builtin(__builtin_amdgcn_s_wait_asynccnt)
  __builtin_amdgcn_s_wait_asynccnt(0);
#else
  asm volatile("s_wait_asynccnt 0x0" ::: "memory");
#endif
}

// D(16x16) = A(16xK, row-major lda) x B^T, where Bm is (16xK, row-major ldb).
// V_WMMA_F32_16X16X4_F32 layouts: A lane m=lane&15, K-pair = 2*(lane>>4);
// B mirrors with lane = n. K must be a multiple of 4.
__device__ __forceinline__ v8f wmma_tile_ABt(const float* A, long lda,
                                             const float* Bm, long ldb,
                                             int K, int lane) {
  const int mn = lane & 15;
  const int h  = lane >> 4;
  const float* ap = A  + (long)mn * lda + 2 * h;
  const float* bp = Bm + (long)mn * ldb + 2 * h;
  v8f acc = {};
  #pragma unroll 4
  for (int k = 0; k < K; k += 4) {
    v2f a = ld2(ap + k);
    v2f b = ld2(bp + k);
    acc = __builtin_amdgcn_wmma_f32_16x16x4_f32(
        false, a, false, b, (short)0, acc, false, false);
  }
  return acc;
}

__global__ void init_kernel(float* d_ot, float* ortho_out) {
  int i = blockIdx.x * blockDim.x + threadIdx.x;
  if (i < B_) d_ot[i] = 0.0f;
  if (i == 0) *ortho_out = 0.0f;
}

__global__ __launch_bounds__(256) void normalize_dirs_kernel(const float* __restrict__ dirs,
                                                             float* __restrict__ dirs_n) {
  __shared__ float red[256];
  const int p = blockIdx.x, tid = threadIdx.x;
  const float* row = dirs + (long)p * D_;
  float s = 0.0f;
  for (int d = tid; d < D_; d += 256) { float v = row[d]; s += v * v; }
  red[tid] = s; __syncthreads();
  for (int o = 128; o; o >>= 1) { if (tid < o) red[tid] += red[tid + o]; __syncthreads(); }
  const float inv = rsqrtf(red[0]);
  for (int d = tid; d < D_; d += 256) dirs_n[(long)p * D_ + d] = row[d] * inv;
}

__global__ __launch_bounds__(256) void rep_kernel(const float* __restrict__ H,
                                                  const int* __restrict__ tt,
                                                  const int* __restrict__ attn,
                                                  float* __restrict__ rep) {
  __shared__ float w0[T_], w1[T_];
  __shared__ int cnt0, cnt1;
  const int b = blockIdx.x, tid = threadIdx.x;
  if (tid == 0) { cnt0 = 0; cnt1 = 0; }
  __syncthreads();
  if (tid < T_) {
    const int t = tt[b * T_ + tid], a = attn[b * T_ + tid];
    const int m0 = (t == 0 && a == 1), m1 = (t == 1 && a == 1);
    w0[tid] = (float)m0; w1[tid] = (float)m1;
    if (m0) atomicAdd(&cnt0, 1);
    if (m1) atomicAdd(&cnt1, 1);
  }
  __syncthreads();
  const float c0 = (float)max(cnt0, 1), c1 = (float)max(cnt1, 1);
  const float* Hb = H + (long)b * T_ * D_;
  for (int d = tid; d < D_; d += 256) {
    float s0 = 0.0f, s1 = 0.0f;
    for (int t = 0; t < T_; ++t) {
      const float v = Hb[(long)t * D_ + d];
      s0 += w0[t] * v; s1 += w1[t] * v;
    }
    rep[(long)b * TWO_D + d]       = Hb[d];            // cls token
    rep[(long)b * TWO_D + D_ + d]  = s0 / c0 - s1 / c1;
  }
}

// proj[b*T+t][p] = sum_d H[row][d] * dirs_n[p][d]; M=B*T, N=P, K=D.
// 8 waves/block cover all 128 p; dirs_n K-chunks are double-buffered into LDS
// with async global->LDS loads (ASYNCcnt), B fragments read via ds_load_b64.
__global__ __launch_bounds__(256) void proj_kernel(const float* __restrict__ H,
                                                   const float* __restrict__ dirs_n,
                                                   float* __restrict__ proj) {
  __shared__ float lbuf[2][P_ * SROW_];
  const int tid  = threadIdx.x;
  const int lane = tid & 31;
  const int wave = tid >> 5;            // 8 waves -> 8 N-tiles
  const long mbase = (long)blockIdx.x * 16;
  const int  nbase = wave * 16;
  const int n = lane & 15, h = lane >> 4;

  // stage layout: thread -> (row = tid/8 + 32*i, 4-float group g = tid%8)
  const int srow = tid >> 3;
  const int sg   = (tid & 7) * 4;

  // prologue: async-stage chunk 0
  #pragma unroll
  for (int i = 0; i < 4; ++i) {
    const int r = srow + i * 32;
    async_copy16(dirs_n + (long)r * D_ + sg, &lbuf[0][r * SROW_ + sg]);
  }
  wait_asynccnt0();
  __syncthreads();

  v8f acc = {};
  const float* ap = H + (mbase + n) * D_ + 2 * h;   // A fragment base (m = lane&15)
  for (int c = 0; c < NCHUNK; ++c) {
    if (c + 1 < NCHUNK) {                            // prefetch next chunk
      const int kb = (c + 1) * KC_;
      float* dst = &lbuf[(c + 1) & 1][0];
      #pragma unroll
      for (int i = 0; i < 4; ++i) {
        const int r = srow + i * 32;
        async_copy16(dirs_n + (long)r * D_ + kb + sg, dst + r * SROW_ + sg);
      }
    }
    const float* lb = &lbuf[c & 1][(nbase + n) * SROW_ + 2 * h];
    const int kg = c * KC_;
    #pragma unroll
    for (int k0 = 0; k0 < KC_; k0 += 4) {
      v2f a = ld2(ap + kg + k0);
      v2f b = ld2(lb + k0);                          // LDS -> ds_load_b64
      acc = __builtin_amdgcn_wmma_f32_16x16x4_f32(
          false, a, false, b, (short)0, acc, false, false);
    }
    wait_asynccnt0();
    __syncthreads();
  }

  #pragma unroll
  for (int r = 0; r < 8; ++r) {
    const long row = mbase + r + 8 * h;
    proj[row * P_ + nbase + n] = acc[r];
  }
}

// fused = concat(C*(1-g), S*g); W rows = [Wc(192); Ws(128)], K = 2D = 1536
__global__ __launch_bounds__(32) void fused_kernel(const float* __restrict__ rep,
                                                   const float* __restrict__ Wc,
                                                   const float* __restrict__ bc,
                                                   const float* __restrict__ Ws,
                                                   const float* __restrict__ bs,
                                                   const float* __restrict__ gate,
                                                   float* __restrict__ fusedo) {
  const int lane  = threadIdx.x & 31;
  const int mbase = blockIdx.x * 16;
  const int nbase = blockIdx.y * 16;
  const int n = lane & 15, h = lane >> 4;
  const int ng = nbase + n;
  const float* wrow = (ng < CD_) ? (Wc + (long)ng * TWO_D)
                                 : (Ws + (long)(ng - CD_) * TWO_D);
  const float* ap = rep + (long)(mbase + n) * TWO_D + 2 * h;  // m == lane&15
  const float* bp = wrow + 2 * h;
  v8f acc = {};
  #pragma unroll 4
  for (int k = 0; k < TWO_D; k += 4) {
    v2f a = ld2(ap + k);
    v2f b = ld2(bp + k);
    acc = __builtin_amdgcn_wmma_f32_16x16x4_f32(
        false, a, false, b, (short)0, acc, false, false);
  }
  const float g     = 1.0f / (1.0f + __expf(-gate[0]));
  const float bias  = (ng < CD_) ? bc[ng] : bs[ng - CD_];
  const float scale = (ng < CD_) ? (1.0f - g) : g;
  #pragma unroll
  for (int r = 0; r < 8; ++r) {
    const int m = mbase + r + 8 * h;
    fusedo[(long)m * NFUSED + ng] = (acc[r] + bias) * scale;
  }
}

// ortho = mean((Wc @ Ws^T)^2), shape 192x128, K=1536
__global__ __launch_bounds__(32) void ortho_kernel(const float* __restrict__ Wc,
                                                   const float* __restrict__ Ws,
                                                   float* __restrict__ ortho_out) {
  const int lane  = threadIdx.x & 31;
  const int mbase = blockIdx.x * 16;
  const int nbase = blockIdx.y * 16;
  v8f acc = wmma_tile_ABt(Wc + (long)mbase * TWO_D, TWO_D,
                          Ws + (long)nbase * TWO_D, TWO_D, TWO_D, lane);
  float s = 0.0f;
  #pragma unroll
  for (int r = 0; r < 8; ++r) s += acc[r] * acc[r];
  for (int off = 16; off; off >>= 1) s += __shfl_down(s, off, 32);
  if (lane == 0) atomicAdd(ortho_out, s * (1.0f / ((float)CD_ * (float)SD_)));
}

// one block per (p, b): bitonic sort masked projections, sliced-W1 distance
__global__ __launch_bounds__(128) void sortdiff_kernel(const float* __restrict__ proj,
                                                       const int* __restrict__ tt,
                                                       const int* __restrict__ attn,
                                                       float* __restrict__ d_ot) {
  __shared__ float xs[T_], ys[T_], red[T_];
  __shared__ int cnt0, cnt1;
  const int p = blockIdx.x, b = blockIdx.y, t = threadIdx.x;
  if (t == 0) { cnt0 = 0; cnt1 = 0; }
  __syncthreads();
  const int tv = tt[b * T_ + t], av = attn[b * T_ + t];
  const int m0 = (tv == 0 && av == 1), m1 = (tv == 1 && av == 1);
  if (m0) atomicAdd(&cnt0, 1);
  if (m1) atomicAdd(&cnt1, 1);
  const float v = proj[((long)b * T_ + t) * P_ + p];
  const float INF = __int_as_float(0x7f800000);
  xs[t] = m0 ? v : INF;
  ys[t] = m1 ? v : INF;
  __syncthreads();
  for (int k = 2; k <= T_; k <<= 1) {
    for (int j = k >> 1; j > 0; j >>= 1) {
      const int ixj = t ^ j;
      if (ixj > t) {
        const bool up = ((t & k) == 0);
        float a = xs[t], c = xs[ixj];
        if ((a > c) == up) { xs[t] = c; xs[ixj] = a; }
        a = ys[t]; c = ys[ixj];
        if ((a > c) == up) { ys[t] = c; ys[ixj] = a; }
      }
      __syncthreads();
    }
  }
  const int mm = min(cnt0, cnt1);
  red[t] = (t < mm) ? fabsf(xs[t] - ys[t]) : 0.0f;
  __syncthreads();
  for (int o = 64; o; o >>= 1) { if (t < o) red[t] += red[t + o]; __syncthreads(); }
  if (t == 0) {
    const float diff = red[0] / (float)max(mm, 1);
    // diff >= 0, so uint bit order == float order
    atomicMax((unsigned int*)&d_ot[b], __float_as_uint(diff));
  }
}

__global__ __launch_bounds__(128) void final_kernel(const float* __restrict__ fusedo,
                                                    const float* __restrict__ d_ot,
                                                    const float* __restrict__ ln_g,
                                                    const float* __restrict__ ln_b,
                                                    const float* __restrict__ Wcls,
                                                    const float* __restrict__ bcls,
                                                    float* __restrict__ logits) {
  __shared__ float rs[128], rq[128], r0[128], r1[128];
  const int b = blockIdx.x, tid = threadIdx.x;
  float vals[3]; int fs[3]; int nf = 0;
  float s = 0.0f, q = 0.0f;
  for (int f = tid; f < FEAT_; f += 128) {
    const float v = (f < NFUSED) ? fusedo[(long)b * NFUSED + f] : d_ot[b];
    vals[nf] = v; fs[nf] = f; ++nf;
    s += v; q += v * v;
  }
  rs[tid] = s; rq[tid] = q; __syncthreads();
  for (int o = 64; o; o >>= 1) {
    if (tid < o) { rs[tid] += rs[tid + o]; rq[tid] += rq[tid + o]; }
    __syncthreads();
  }
  const float mu   = rs[0] / (float)FEAT_;
  const float var  = rq[0] / (float)FEAT_ - mu * mu;
  const float rstd = rsqrtf(var + 1e-5f);
  float a0 = 0.0f, a1 = 0.0f;
  for (int i = 0; i < nf; ++i) {
    const int f = fs[i];
    const float nv = (vals[i] - mu) * rstd * ln_g[f] + ln_b[f];
    a0 += nv * Wcls[f];
    a1 += nv * Wcls[FEAT_ + f];
  }
  r0[tid] = a0; r1[tid] = a1; __syncthreads();
  for (int o = 64; o; o >>= 1) {
    if (tid < o) { r0[tid] += r0[tid + o]; r1[tid] += r1[tid + o]; }
    __syncthreads();
  }
  if (tid == 0) {
    logits[b * 2 + 0] = r0[0] + bcls[0];
    logits[b * 2 + 1] = r1[0] + bcls[1];
  }
}

extern "C" void kernel_launch(void* const* d_in, const int* in_sizes, int n_in,
                              void* d_out, int out_size, void* d_ws, size_t ws_size,
                              hipStream_t stream) {
  const float* H    = (const float*)d_in[0];
  const int*   tt   = (const int*)d_in[1];
  const int*   attn = (const int*)d_in[2];
  const float* Wc   = (const float*)d_in[3];
  const float* bc   = (const float*)d_in[4];
  const float* Ws   = (const float*)d_in[5];
  const float* bs   = (const float*)d_in[6];
  const float* gate = (const float*)d_in[7];
  const float* ln_g = (const float*)d_in[8];
  const float* ln_b = (const float*)d_in[9];
  const float* Wcls = (const float*)d_in[10];
  const float* bcls = (const float*)d_in[11];
  const float* dirs = (const float*)d_in[12];

  float* logits = (float*)d_out;          // 512*2
  float* ortho  = (float*)d_out + B_ * 2; // 1

  float* ws     = (float*)d_ws;
  float* dirs_n = ws;                                   // 128*768
  float* rep    = dirs_n + (size_t)P_ * D_;             // 512*1536
  float* fusedb = rep    + (size_t)B_ * TWO_D;          // 512*320
  float* proj   = fusedb + (size_t)B_ * NFUSED;         // 512*128*128
  float* d_ot   = proj   + (size_t)B_ * T_ * P_;        // 512

  init_kernel<<<dim3(2), dim3(256), 0, stream>>>(d_ot, ortho);
  normalize_dirs_kernel<<<dim3(P_), dim3(256), 0, stream>>>(dirs, dirs_n);
  rep_kernel<<<dim3(B_), dim3(256), 0, stream>>>(H, tt, attn, rep);
  proj_kernel<<<dim3((B_ * T_) / 16), dim3(256), 0, stream>>>(H, dirs_n, proj);
  fused_kernel<<<dim3(B_ / 16, NFUSED / 16), dim3(32), 0, stream>>>(rep, Wc, bc, Ws, bs, gate, fusedb);
  ortho_kernel<<<dim3(CD_ / 16, SD_ / 16), dim3(32), 0, stream>>>(Wc, Ws, ortho);
  sortdiff_kernel<<<dim3(P_, B_), dim3(T_), 0, stream>>>(proj, tt, attn, d_ot);
  final_kernel<<<dim3(B_), dim3(128), 0, stream>>>(fusedb, d_ot, ln_g, ln_b, Wcls, bcls, logits);
}